// PerFrameAttentionPooling_28913719837042
// MI455X (gfx1250) — compile-verified
//
#include <hip/hip_runtime.h>

typedef __attribute__((ext_vector_type(2))) float v2f;
typedef __attribute__((ext_vector_type(8))) float v8f;

// Only meaningful during the device pass; host pass reports 0 for amdgcn builtins.
#if defined(__HIP_DEVICE_COMPILE__)
#if !__has_builtin(__builtin_amdgcn_wmma_f32_16x16x4_f32)
#error "wmma_f32_16x16x4_f32 builtin not available on this toolchain (device pass)"
#endif
#endif

constexpr int DIM    = 1024;
constexpr int NHEADS = 16;
constexpr int HDIM   = 64;
constexpr int SEQ    = 576;   // H*W = 24*24
constexpr int FRAMES = 128;   // B*T = 8*16
constexpr float QSCALE = 0.125f;  // 1/sqrt(64)
constexpr float EPS = 1e-6f;

// workspace layout (float offsets)
constexpr int OFF_QT  = 0;                       // qt transposed [h][d], 16*1024
constexpr int OFF_SB  = OFF_QT + NHEADS * DIM;   // 16
constexpr int OFF_QSC = OFF_SB + NHEADS;         // scaled q, 1024
constexpr int OFF_PX  = OFF_QSC + DIM;           // pooled x [frame][d][h], 128*1024*16
constexpr int OFF_O   = OFF_PX + FRAMES * DIM * NHEADS;  // attn output rows, 128*1024

// LDS padded strides (64 banks, avoid stride%64==0)
constexpr int QT_STRIDE = DIM + 8;   // 1032
constexpr int AT_STRIDE = SEQ + 8;   // 584

__device__ __forceinline__ float wave_sum(float v) {
  #pragma unroll
  for (int off = 16; off; off >>= 1) v += __shfl_xor(v, off, 32);
  return v;
}
__device__ __forceinline__ float wave_max(float v) {
  #pragma unroll
  for (int off = 16; off; off >>= 1) v = fmaxf(v, __shfl_xor(v, off, 32));
  return v;
}

// ---------------- K0a: q = probe @ Wq^T + bq, scaled. One wave per output d.
__global__ void k_q(const float* __restrict__ probe, const float* __restrict__ wq,
                    const float* __restrict__ inb, float* __restrict__ ws) {
  int w = threadIdx.x >> 5, lane = threadIdx.x & 31;
  int d = blockIdx.x * 8 + w;                 // 128 blocks * 8 waves = 1024
  const float* row = wq + (size_t)d * DIM;
  float acc = 0.f;
  for (int j = lane; j < DIM; j += 32) acc += probe[j] * row[j];
  acc = wave_sum(acc);
  if (lane == 0) ws[OFF_QSC + d] = (acc + inb[d]) * QSCALE;
}

// ---------------- K0b: qt[h][d] = sum_j Wk[h*64+j][d]*qsc[h*64+j]; sb[h]=qh.bk_h
__global__ void k_qt(const float* __restrict__ wk, const float* __restrict__ inb,
                     float* __restrict__ ws) {
  int g = blockIdx.x * blockDim.x + threadIdx.x;   // 64*256 = 16384
  int h = g >> 10, d = g & (DIM - 1);
  const float* qsc = ws + OFF_QSC;
  float acc = 0.f;
  for (int j = 0; j < HDIM; ++j)
    acc += wk[(size_t)(h * HDIM + j) * DIM + d] * qsc[h * HDIM + j];
  ws[OFF_QT + h * DIM + d] = acc;
  if (g < NHEADS) {
    float s = 0.f;
    for (int j = 0; j < HDIM; ++j) s += qsc[g * HDIM + j] * inb[DIM + g * HDIM + j];
    ws[OFF_SB + g] = s;
  }
}

// ---------------- K1: per-frame fused scores -> softmax -> weighted pooling (px)
// grid = 128 frames, block = 512 (16 waves), dyn LDS = qt(16*1032) + attn(16*584)
__global__ void k_attend(const float* __restrict__ x, float* ws) {
  extern __shared__ float smem[];
  float* qt_s   = smem;                      // [n*QT_STRIDE + k]
  float* attn_s = smem + NHEADS * QT_STRIDE; // [h*AT_STRIDE + s]
  const int tid = threadIdx.x, w = tid >> 5, lane = tid & 31;
  const int frame = blockIdx.x;
  const int n = lane & 15;
  const int koff = (lane >> 4) << 1;         // lanes 16-31 handle K+2,K+3
  const float* xf = x + (size_t)frame * SEQ * DIM;

  for (int i = tid; i < NHEADS * DIM; i += 512)
    qt_s[(i >> 10) * QT_STRIDE + (i & (DIM - 1))] = ws[OFF_QT + i];
  __syncthreads();

  // --- scores: X(16xK) @ qt(Kx16) per token tile, C preloaded with sb[h]
  const float sbv = ws[OFF_SB + n];
  for (int t = w; t < SEQ / 16; t += 16) {
    v8f c;
    #pragma unroll
    for (int r = 0; r < 8; ++r) c[r] = sbv;
    const float* arow = xf + (size_t)(t * 16 + n) * DIM + koff;
    const float* brow = qt_s + n * QT_STRIDE + koff;
    for (int k0 = 0; k0 < DIM; k0 += 4) {
      v2f a = *(const v2f*)(arow + k0);
      v2f b = *(const v2f*)(brow + k0);
      c = __builtin_amdgcn_wmma_f32_16x16x4_f32(false, a, false, b, (short)0, c, false, false);
    }
    const int mbase = t * 16 + ((lane >> 4) << 3);
    #pragma unroll
    for (int r = 0; r < 8; ++r) attn_s[n * AT_STRIDE + mbase + r] = c[r];
  }
  __syncthreads();

  // --- softmax over S: wave w handles head w
  {
    float* row = attn_s + w * AT_STRIDE;
    float mx = -3.4e38f;
    for (int s = lane; s < SEQ; s += 32) mx = fmaxf(mx, row[s]);
    mx = wave_max(mx);
    float sum = 0.f;
    for (int s = lane; s < SEQ; s += 32) {
      float e = __expf(row[s] - mx);
      row[s] = e;
      sum += e;
    }
    sum = wave_sum(sum);
    const float rinv = 1.f / sum;
    for (int s = lane; s < SEQ; s += 32) row[s] *= rinv;
  }
  __syncthreads();

  // --- px[d][h] = sum_s x[s][d] * attn[h][s] : (X^T) @ attn^T, 4 d-tiles per wave
  #pragma unroll
  for (int i = 0; i < 4; ++i) {
    const int dt = w * 4 + i;
    const int d = dt * 16 + n;
    const float* ac = xf + d + (size_t)koff * DIM;
    const float* br = attn_s + n * AT_STRIDE + koff;
    v8f c = {};
    for (int s0 = 0; s0 < SEQ; s0 += 4) {
      v2f a;
      a.x = ac[(size_t)s0 * DIM];
      a.y = ac[(size_t)(s0 + 1) * DIM];
      v2f b = *(const v2f*)(br + s0);
      c = __builtin_amdgcn_wmma_f32_16x16x4_f32(false, a, false, b, (short)0, c, false, false);
    }
    const int dbase = dt * 16 + ((lane >> 4) << 3);
    #pragma unroll
    for (int r = 0; r < 8; ++r)
      ws[OFF_PX + (size_t)(frame * DIM + dbase + r) * NHEADS + n] = c[r];
  }
}

// ---------------- K2: per-head V projection: O_h(128x64) = PX_h(128x1024) @ Wv_h^T + bv
// 16 heads * 8 m-tiles * 4 n-tiles = 512 wave-tiles -> 64 blocks * 8 waves
__global__ void k_vproj(const float* __restrict__ wv, const float* __restrict__ bv,
                        float* ws) {
  const int w = threadIdx.x >> 5, lane = threadIdx.x & 31;
  const int id = blockIdx.x * 8 + w;
  const int h = id >> 5, rem = id & 31, mt = rem >> 2, nt = rem & 3;
  const int n = lane & 15, koff = (lane >> 4) << 1;
  const int f = mt * 16 + n;
  const float* ap = ws + OFF_PX + (size_t)(f * DIM + koff) * NHEADS + h;
  const float* bp = wv + (size_t)(h * HDIM + nt * 16 + n) * DIM + koff;
  v8f c = {};
  for (int k0 = 0; k0 < DIM; k0 += 4) {
    v2f a;
    a.x = ap[(size_t)k0 * NHEADS];
    a.y = ap[(size_t)(k0 + 1) * NHEADS];
    v2f b = *(const v2f*)(bp + k0);
    c = __builtin_amdgcn_wmma_f32_16x16x4_f32(false, a, false, b, (short)0, c, false, false);
  }
  const int col = h * HDIM + nt * 16 + n;
  const float bias = bv[col];
  const int mbase = mt * 16 + ((lane >> 4) << 3);
  #pragma unroll
  for (int r = 0; r < 8; ++r)
    ws[OFF_O + (size_t)(mbase + r) * DIM + col] = c[r] + bias;
}

// ---------------- K3: out-projection GEMM (128x1024 @ 1024x1024^T) fused with LayerNorm
// grid = 8 blocks (16 frames each), block = 512 (16 waves, 4 n-tiles per wave)
__global__ void k_out_ln(const float* __restrict__ ow, const float* __restrict__ ob,
                         const float* __restrict__ lg, const float* __restrict__ lb,
                         const float* __restrict__ ws, float* __restrict__ out) {
  extern __shared__ float rowbuf[];          // 16 * 1024
  const int tid = threadIdx.x, w = tid >> 5, lane = tid & 31;
  const int f0 = blockIdx.x * 16;
  const int n = lane & 15, koff = (lane >> 4) << 1;

  const float* ap = ws + OFF_O + (size_t)(f0 + n) * DIM + koff;
  const float* bp[4];
  #pragma unroll
  for (int i = 0; i < 4; ++i)
    bp[i] = ow + (size_t)((w * 4 + i) * 16 + n) * DIM + koff;

  v8f acc[4] = {};
  for (int k0 = 0; k0 < DIM; k0 += 4) {
    v2f a = *(const v2f*)(ap + k0);
    #pragma unroll
    for (int i = 0; i < 4; ++i) {
      v2f b = *(const v2f*)(bp[i] + k0);
      acc[i] = __builtin_amdgcn_wmma_f32_16x16x4_f32(false, a, false, b, (short)0, acc[i], false, false);
    }
  }
  const int mbase = (lane >> 4) << 3;
  #pragma unroll
  for (int i = 0; i < 4; ++i) {
    const int col = (w * 4 + i) * 16 + n;
    const float bias = ob[col];
    #pragma unroll
    for (int r = 0; r < 8; ++r)
      rowbuf[(mbase + r) * DIM + col] = acc[i][r] + bias;
  }
  __syncthreads();

  // LayerNorm: wave w normalizes row w
  const float* row = rowbuf + w * DIM;
  float s = 0.f, s2 = 0.f;
  for (int i = lane; i < DIM; i += 32) {
    float v = row[i];
    s += v;
    s2 += v * v;
  }
  s = wave_sum(s);
  s2 = wave_sum(s2);
  const float mean = s * (1.f / DIM);
  const float var = s2 * (1.f / DIM) - mean * mean;
  const float rstd = rsqrtf(var + EPS);
  float* orow = out + (size_t)(f0 + w) * DIM;
  for (int i = lane; i < DIM; i += 32)
    orow[i] = (row[i] - mean) * rstd * lg[i] + lb[i];
}

extern "C" void kernel_launch(void* const* d_in, const int* in_sizes, int n_in,
                              void* d_out, int out_size, void* d_ws, size_t ws_size,
                              hipStream_t stream) {
  const float* x     = (const float*)d_in[0];
  const float* probe = (const float*)d_in[1];
  const float* inw   = (const float*)d_in[2];   // [3*D, D]: Wq | Wk | Wv
  const float* inb   = (const float*)d_in[3];   // [3*D]
  const float* ow    = (const float*)d_in[4];
  const float* ob    = (const float*)d_in[5];
  const float* lg    = (const float*)d_in[6];
  const float* lb    = (const float*)d_in[7];
  float* ws  = (float*)d_ws;
  float* out = (float*)d_out;

  k_q  <<<128, 256, 0, stream>>>(probe, inw, inb, ws);
  k_qt <<<64, 256, 0, stream>>>(inw + (size_t)DIM * DIM, inb, ws);
  k_attend<<<FRAMES, 512,
             (NHEADS * QT_STRIDE + NHEADS * AT_STRIDE) * sizeof(float), stream>>>(x, ws);
  k_vproj<<<64, 256, 0, stream>>>(inw + 2 * (size_t)DIM * DIM, inb + 2 * DIM, ws);
  k_out_ln<<<8, 512, NHEADS * DIM * sizeof(float), stream>>>(ow, ob, lg, lb, ws, out);
}